// GCNLayer_63943473103420
// MI455X (gfx1250) — compile-verified
//
#include <hip/hip_runtime.h>

typedef __attribute__((ext_vector_type(2))) float v2f;
typedef __attribute__((ext_vector_type(8))) float v8f;

#define CH 64

// ---------------------------------------------------------------------------
// Degree kernels (computed once; edges identical for both layers)
// ---------------------------------------------------------------------------
__global__ __launch_bounds__(256) void zero_f32(float* __restrict__ p, int n) {
    int i = blockIdx.x * 256 + threadIdx.x;
    if (i < n) p[i] = 0.0f;
}

__global__ __launch_bounds__(256) void count_deg(const long long* __restrict__ dst,
                                                 float* __restrict__ deg, int n_edges) {
    int i = blockIdx.x * 256 + threadIdx.x;
    if (i < n_edges) atomicAdd(&deg[dst[i]], 1.0f);
}

__global__ __launch_bounds__(256) void invert_deg(float* __restrict__ deg, int n) {
    int i = blockIdx.x * 256 + threadIdx.x;
    if (i < n) deg[i] = 1.0f / fmaxf(deg[i], 1.0f);
}

// ---------------------------------------------------------------------------
// Triple GEMM: h0 = x@w0, h1 = x@w1, agg = x@root + bias
// One wave computes a 16x16 tile of all three outputs using
// V_WMMA_F32_16X16X4_F32 (fp32 CDNA5 matrix op), 16 K-steps (K=64).
//
// A layout (16x4 f32): lane<16 -> M=lane, VGPR0=K+0,VGPR1=K+1;
//                      lane>=16 -> M=lane-16, VGPR0=K+2,VGPR1=K+3.
// B layout (4x16 f32): lane<16 -> N=lane, K+0/K+1; lane>=16 -> N=lane-16, K+2/K+3.
// C/D (16x16 f32, 8 VGPRs): VGPR r: lanes 0-15 M=r, lanes 16-31 M=r+8; N=lane&15.
// ---------------------------------------------------------------------------
__global__ __launch_bounds__(256) void spline_gemm3(
    const float* __restrict__ x,     // n x 64
    const float* __restrict__ w0,    // 64 x 64
    const float* __restrict__ w1,    // 64 x 64
    const float* __restrict__ root,  // 64 x 64
    const float* __restrict__ bias,  // 64
    float* __restrict__ h0,
    float* __restrict__ h1,
    float* __restrict__ agg,         // gets x@root + bias
    int n_nodes)
{
    const int lane = threadIdx.x & 31;
    const int wave = threadIdx.x >> 5;
    const int gw   = blockIdx.x * 8 + wave;          // global wave index
    const int rowblks = (n_nodes + 15) >> 4;
    const int rowblk  = gw >> 2;                     // wave-uniform
    const int ntile   = gw & 3;                      // which 16-col tile
    if (rowblk >= rowblks) return;                   // wave-uniform: EXEC stays all-1s

    const int half = lane >> 4;                      // 0 or 1
    const int l15  = lane & 15;
    const int row  = rowblk * 16 + l15;              // A-fragment row (M)
    const int kofs = half * 2;                       // K sub-offset for this half-wave
    const int col  = ntile * 16 + l15;               // B/C column (N)

    v8f c0 = {};
    v8f c1 = {};
    v8f cr = {};

    const float* xrow = x + (size_t)row * CH;

#pragma unroll
    for (int kk = 0; kk < 16; ++kk) {
        const int k = kk * 4;

        // A fragment: 2 consecutive fp32 of this lane's row (8B aligned)
        v2f a = *(const v2f*)(xrow + k + kofs);

        // B fragments: column `col` of rows k+kofs, k+kofs+1 of each weight
        v2f b0, b1, br;
        b0.x = w0[(k + kofs) * CH + col];
        b0.y = w0[(k + kofs + 1) * CH + col];
        b1.x = w1[(k + kofs) * CH + col];
        b1.y = w1[(k + kofs + 1) * CH + col];
        br.x = root[(k + kofs) * CH + col];
        br.y = root[(k + kofs + 1) * CH + col];

        c0 = __builtin_amdgcn_wmma_f32_16x16x4_f32(false, a, false, b0, (short)0, c0, false, false);
        c1 = __builtin_amdgcn_wmma_f32_16x16x4_f32(false, a, false, b1, (short)0, c1, false, false);
        cr = __builtin_amdgcn_wmma_f32_16x16x4_f32(false, a, false, br, (short)0, cr, false, false);
    }

    const float bn = bias[col];

#pragma unroll
    for (int r = 0; r < 8; ++r) {
        const int m = rowblk * 16 + half * 8 + r;
        const size_t o = (size_t)m * CH + col;
        h0[o]  = c0[r];
        h1[o]  = c1[r];
        agg[o] = cr[r] + bn;
    }
}

// ---------------------------------------------------------------------------
// Edge scatter: agg[dst] += ((1-u)*h0[src] + u*h1[src]) * invdeg[dst]
// 16 threads per edge, float4 per thread (vector gather + 4-wide f32 atomics).
// h0/h1/agg are 25.6 MB each -> L2-resident on MI455X (192 MB L2).
// ---------------------------------------------------------------------------
__global__ __launch_bounds__(256) void spline_scatter(
    const float* __restrict__ h0, const float* __restrict__ h1,
    const long long* __restrict__ src, const long long* __restrict__ dst,
    const float* __restrict__ u, const float* __restrict__ invdeg,
    float* __restrict__ agg, int n_edges)
{
    int tid = blockIdx.x * 256 + threadIdx.x;
    int e = tid >> 4;
    if (e >= n_edges) return;
    int c = (tid & 15) << 2;

    long long s = src[e];
    long long d = dst[e];
    float uu  = u[e];
    float idg = invdeg[d];
    float a0 = (1.0f - uu) * idg;
    float a1 = uu * idg;

    float4 q0 = *(const float4*)(h0 + (size_t)s * CH + c);
    float4 q1 = *(const float4*)(h1 + (size_t)s * CH + c);

    float* o = agg + (size_t)d * CH + c;
    atomicAdd(o + 0, a0 * q0.x + a1 * q1.x);
    atomicAdd(o + 1, a0 * q0.y + a1 * q1.y);
    atomicAdd(o + 2, a0 * q0.z + a1 * q1.z);
    atomicAdd(o + 3, a0 * q0.w + a1 * q1.w);
}

// ---------------------------------------------------------------------------
// Launch: deg once, then (gemm3 -> scatter) x 2 layers.
// ---------------------------------------------------------------------------
extern "C" void kernel_launch(void* const* d_in, const int* in_sizes, int n_in,
                              void* d_out, int out_size, void* d_ws, size_t ws_size,
                              hipStream_t stream)
{
    const float*     x      = (const float*)d_in[0];
    const long long* eidx   = (const long long*)d_in[1];   // (2, E) int64
    const float*     eattr  = (const float*)d_in[2];       // (E, 1)
    const float*     w1     = (const float*)d_in[3];       // (2, 64, 64)
    const float*     root1  = (const float*)d_in[4];
    const float*     b1     = (const float*)d_in[5];
    const float*     w2     = (const float*)d_in[6];
    const float*     root2  = (const float*)d_in[7];
    const float*     b2     = (const float*)d_in[8];
    float*           out    = (float*)d_out;

    const int N = in_sizes[0] / CH;
    const int E = in_sizes[2];

    const long long* src = eidx;
    const long long* dst = eidx + (size_t)E;

    // workspace: h0 | h1 | y1 | invdeg
    float* h0 = (float*)d_ws;
    float* h1 = h0 + (size_t)N * CH;
    float* y1 = h1 + (size_t)N * CH;
    float* invdeg = y1 + (size_t)N * CH;

    const int nThreads = 256;
    const int nBlocksN = (N + nThreads - 1) / nThreads;
    const int nBlocksE = (E + nThreads - 1) / nThreads;

    // ---- degree (shared by both layers) ----
    zero_f32<<<nBlocksN, nThreads, 0, stream>>>(invdeg, N);
    count_deg<<<nBlocksE, nThreads, 0, stream>>>(dst, invdeg, E);
    invert_deg<<<nBlocksN, nThreads, 0, stream>>>(invdeg, N);

    // GEMM grid: one wave per 16x16 tile, 4 col-tiles, 8 waves per block
    const int rowblks = (N + 15) / 16;
    const int gemmWaves = rowblks * 4;
    const int gemmBlocks = (gemmWaves + 7) / 8;

    // Scatter grid: 16 threads per edge
    const long long scatThreads = (long long)E * 16;
    const int scatBlocks = (int)((scatThreads + nThreads - 1) / nThreads);

    // ---- layer 1 ----
    spline_gemm3<<<gemmBlocks, nThreads, 0, stream>>>(
        x, w1, w1 + CH * CH, root1, b1, h0, h1, y1, N);
    spline_scatter<<<scatBlocks, nThreads, 0, stream>>>(
        h0, h1, src, dst, eattr, invdeg, y1, E);

    // ---- layer 2 (writes final result into d_out) ----
    spline_gemm3<<<gemmBlocks, nThreads, 0, stream>>>(
        y1, w2, w2 + CH * CH, root2, b2, h0, h1, out, N);
    spline_scatter<<<scatBlocks, nThreads, 0, stream>>>(
        h0, h1, src, dst, eattr, invdeg, out, E);
}